// SimpleCornerGNN_35880156790903
// MI455X (gfx1250) — compile-verified
//
#include <hip/hip_runtime.h>
#include <hip/hip_bf16.h>
#include <stdint.h>

#define N_NODES 100000
#define N_EDGES 1600000
#define HDIM    64

typedef float v2f __attribute__((ext_vector_type(2)));
typedef float v8f __attribute__((ext_vector_type(8)));

// ---------------- degree / normalization ----------------

__global__ void k_init_deg(float* deg) {
  int i = blockIdx.x * blockDim.x + threadIdx.x;
  if (i < N_NODES) deg[i] = 1.0f;  // self-loop contributes 1 to every node
}

__global__ void k_deg_scatter(const long long* __restrict__ dst, float* deg) {
  int e = blockIdx.x * blockDim.x + threadIdx.x;
  if (e < N_EDGES) atomicAdd(&deg[(int)dst[e]], 1.0f);
}

__global__ void k_rsqrt_inplace(float* deg) {
  int i = blockIdx.x * blockDim.x + threadIdx.x;
  if (i < N_NODES) deg[i] = rsqrtf(deg[i]);  // deg >= 1 always (self-loop)
}

// ---------------- layer 1 linear: x[N,3] @ W1[3,64] ----------------

__global__ void k_lin3(const float* __restrict__ x, const float* __restrict__ W1,
                       float* __restrict__ out) {
  int gid = blockIdx.x * blockDim.x + threadIdx.x;
  if (gid >= N_NODES * HDIM) return;
  int n = gid >> 6, f = gid & 63;
  const float* xr = x + n * 3;
  out[gid] = fmaf(xr[0], W1[f], fmaf(xr[1], W1[64 + f], xr[2] * W1[128 + f]));
}

// ---------------- layers 2/3 linear: H[N,64] @ W[64,64] via WMMA f32 ----------------
// One wave -> one 16x16 output tile. Block = 4 waves -> 16 rows x 64 cols.
// A 16x4 f32 layout : lanes 0-15 rows M=lane, VGPR{0,1}=K{0,1}; lanes 16-31 K{2,3}.
// B 4x16 f32 layout : lanes 0-15 cols N=lane, VGPR{0,1}=K{0,1}; lanes 16-31 K{2,3}.
// C/D 16x16 f32     : VGPR v = row (v + 8*(lane>>4)), col = lane&15.

__global__ __launch_bounds__(128) void k_gemm64(const float* __restrict__ A,
                                                const float* __restrict__ W,
                                                float* __restrict__ out) {
  const int lane  = threadIdx.x & 31;
  const int ncol0 = (threadIdx.x >> 5) * 16;  // 0,16,32,48
  const int row0  = blockIdx.x * 16;          // 6250 blocks, exact (100000 = 16*6250)
  const int m     = lane & 15;
  const int kb    = (lane >> 4) * 2;
  const int nn    = lane & 15;

  v8f c = {};
#if __has_builtin(__builtin_amdgcn_wmma_f32_16x16x4_f32)
  const float* arow = A + (size_t)(row0 + m) * HDIM;
  #pragma unroll
  for (int k0 = 0; k0 < HDIM; k0 += 4) {
    v2f a, b;
    a.x = arow[k0 + kb];
    a.y = arow[k0 + kb + 1];
    b.x = W[(k0 + kb) * HDIM + ncol0 + nn];
    b.y = W[(k0 + kb + 1) * HDIM + ncol0 + nn];
    c = __builtin_amdgcn_wmma_f32_16x16x4_f32(false, a, false, b,
                                              (short)0, c, false, false);
  }
#else
  // Scalar fallback producing the same C layout (should not be taken on gfx1250).
  #pragma unroll
  for (int v = 0; v < 8; ++v) {
    int mr = (lane >> 4) * 8 + v;
    float acc = 0.f;
    for (int k = 0; k < HDIM; ++k)
      acc = fmaf(A[(size_t)(row0 + mr) * HDIM + k], W[k * HDIM + ncol0 + nn], acc);
    c[v] = acc;
  }
#endif

  const int mbase = (lane >> 4) * 8;
  #pragma unroll
  for (int v = 0; v < 8; ++v)
    out[(size_t)(row0 + mbase + v) * HDIM + ncol0 + nn] = c[v];
}

// ---------------- zero accumulator ----------------

__global__ void k_zero(float* p, int count) {
  int i = blockIdx.x * blockDim.x + threadIdx.x;
  if (i < count) p[i] = 0.0f;
}

// ---------------- edge scatter: acc[dst] += lin[src] * dinv[src]*dinv[dst] ----------------
// One thread per (edge, 4-feature chunk): float4 gather + 4 f32 atomics.

__global__ void k_edge_scatter(const long long* __restrict__ src,
                               const long long* __restrict__ dst,
                               const float* __restrict__ dinv,
                               const float* __restrict__ lin,
                               float* acc) {
  int gid = blockIdx.x * blockDim.x + threadIdx.x;
  if (gid >= N_EDGES * 16) return;
  int e  = gid >> 4;
  int f4 = (gid & 15) << 2;
  int s = (int)src[e], d = (int)dst[e];
  float w = dinv[s] * dinv[d];
  const float4 v = *reinterpret_cast<const float4*>(lin + (size_t)s * HDIM + f4);
  float* ap = acc + (size_t)d * HDIM + f4;
  atomicAdd(ap + 0, v.x * w);
  atomicAdd(ap + 1, v.y * w);
  atomicAdd(ap + 2, v.z * w);
  atomicAdd(ap + 3, v.w * w);
}

// ---------------- finalize: acc += self-loop + bias, optional ReLU (in place) ----------------

template <bool RELU>
__global__ void k_finalize(const float* __restrict__ lin, const float* __restrict__ dinv,
                           const float* __restrict__ bias, float* acc) {
  int gid = blockIdx.x * blockDim.x + threadIdx.x;
  if (gid >= N_NODES * HDIM) return;
  int n = gid >> 6, f = gid & 63;
  float di = dinv[n];
  float v = acc[gid] + lin[gid] * di * di + bias[f];
  if (RELU) v = fmaxf(v, 0.0f);
  acc[gid] = v;
}

// ---------------- head: logits[n] = dot(h3[n], Wh) + bh ----------------

__global__ void k_head(const float* __restrict__ h, const float* __restrict__ Wh,
                       const float* __restrict__ bh, float* __restrict__ out) {
  int t = blockIdx.x * blockDim.x + threadIdx.x;
  int n = t >> 5;           // wave32 per node
  int lane = t & 31;
  if (n >= N_NODES) return;
  const float* hr = h + (size_t)n * HDIM;
  float v = fmaf(hr[lane], Wh[lane], hr[lane + 32] * Wh[lane + 32]);
  #pragma unroll
  for (int off = 16; off; off >>= 1) v += __shfl_xor(v, off, 32);
  if (lane == 0) out[n] = v + bh[0];
}

// ---------------- driver ----------------

extern "C" void kernel_launch(void* const* d_in, const int* in_sizes, int n_in,
                              void* d_out, int out_size, void* d_ws, size_t ws_size,
                              hipStream_t stream) {
  const float*     x   = (const float*)d_in[0];
  const long long* ei  = (const long long*)d_in[1];
  const float*     W1  = (const float*)d_in[2];
  const float*     b1  = (const float*)d_in[3];
  const float*     W2  = (const float*)d_in[4];
  const float*     b2  = (const float*)d_in[5];
  const float*     W3  = (const float*)d_in[6];
  const float*     b3  = (const float*)d_in[7];
  const float*     Wh  = (const float*)d_in[8];
  const float*     bh  = (const float*)d_in[9];
  const long long* src = ei;
  const long long* dst = ei + N_EDGES;

  float* dinv = (float*)d_ws;
  float* buf0 = dinv + ((N_NODES + 255) & ~255);
  float* buf1 = buf0 + (size_t)N_NODES * HDIM;
  float* buf2 = buf1 + (size_t)N_NODES * HDIM;
  float* logits = (float*)d_out;

  const int NF = N_NODES * HDIM;              // 6.4M
  const int TB = 256;
  const int gN   = (N_NODES + TB - 1) / TB;
  const int gE   = (N_EDGES + TB - 1) / TB;
  const int gNF  = (NF + TB - 1) / TB;
  const int gE16 = (N_EDGES * 16 + TB - 1) / TB;
  const int gGemm = N_NODES / 16;             // 6250, exact
  const int gHead = (N_NODES * 32 + TB - 1) / TB;

  // normalization: deg -> dinv (in place)
  k_init_deg<<<gN, TB, 0, stream>>>(dinv);
  k_deg_scatter<<<gE, TB, 0, stream>>>(dst, dinv);
  k_rsqrt_inplace<<<gN, TB, 0, stream>>>(dinv);

  // layer 1: lin -> buf0, acc -> buf1 (becomes h1)
  k_lin3<<<gNF, TB, 0, stream>>>(x, W1, buf0);
  k_zero<<<gNF, TB, 0, stream>>>(buf1, NF);
  k_edge_scatter<<<gE16, TB, 0, stream>>>(src, dst, dinv, buf0, buf1);
  k_finalize<true><<<gNF, TB, 0, stream>>>(buf0, dinv, b1, buf1);

  // layer 2: in=buf1, lin -> buf0 (WMMA), acc -> buf2 (becomes h2)
  k_gemm64<<<gGemm, 128, 0, stream>>>(buf1, W2, buf0);
  k_zero<<<gNF, TB, 0, stream>>>(buf2, NF);
  k_edge_scatter<<<gE16, TB, 0, stream>>>(src, dst, dinv, buf0, buf2);
  k_finalize<true><<<gNF, TB, 0, stream>>>(buf0, dinv, b2, buf2);

  // layer 3: in=buf2, lin -> buf0 (WMMA), acc -> buf1 (becomes h3), no ReLU
  k_gemm64<<<gGemm, 128, 0, stream>>>(buf2, W3, buf0);
  k_zero<<<gNF, TB, 0, stream>>>(buf1, NF);
  k_edge_scatter<<<gE16, TB, 0, stream>>>(src, dst, dinv, buf0, buf1);
  k_finalize<false><<<gNF, TB, 0, stream>>>(buf0, dinv, b3, buf1);

  // head
  k_head<<<gHead, TB, 0, stream>>>(buf1, Wh, bh, logits);
}